// NNUEModel_49160195670625
// MI455X (gfx1250) — compile-verified
//
#include <hip/hip_runtime.h>
#include <hip/hip_bf16.h>
#include <math.h>

typedef __attribute__((ext_vector_type(2))) float v2f;
typedef __attribute__((ext_vector_type(4))) float v4f;
typedef __attribute__((ext_vector_type(8))) float v8f;

#define EMB_DIM 256
#define N_FEAT 32
#define ROWS 32          // batch rows per workgroup
#define ACC_STRIDE 260   // 256 + 4 pad: column reads hit 16 distinct banks; rows stay 16B-aligned
#define H_STRIDE 33      // 32 + 1 pad for conflict-free row-wise reads

__global__ __launch_bounds__(256) void nnue_fused_kernel(
    const int* __restrict__ indices, const float* __restrict__ emb,
    const float* __restrict__ w1, const float* __restrict__ b1,
    const float* __restrict__ w2, const float* __restrict__ b2,
    const float* __restrict__ w3, const float* __restrict__ b3,
    float* __restrict__ out, int batch)
{
    __shared__ float sh_acc[ROWS * ACC_STRIDE];  // 33,280 B
    __shared__ float sh_h1 [ROWS * H_STRIDE];    //  4,224 B
    __shared__ float sh_h2 [ROWS * H_STRIDE];    //  4,224 B

    const int tid  = threadIdx.x;
    const int lane = tid & 31;
    const int wave = tid >> 5;
    const int row0 = blockIdx.x * ROWS;

    // ---------------- Phase 1: gather + feature-sum (bandwidth-dominant) ----------------
    // Each wave processes 4 batch rows; lane l owns embedding dims [8l, 8l+8).
    // The row's 32 feature indices are fetched with ONE vector load (lane == feature id)
    // and broadcast with __shfl, turning 32 scalar loads into 1 vmem op.
    for (int rr = 0; rr < 4; ++rr) {
        const int rl = wave * 4 + rr;
        const int b  = row0 + rl;
        const int myidx = indices[lane * batch + b];   // indices[f][b], f = lane
        v4f a0 = {0.f, 0.f, 0.f, 0.f};
        v4f a1 = {0.f, 0.f, 0.f, 0.f};
        #pragma unroll 8
        for (int f = 0; f < N_FEAT; ++f) {
            const int idx = __shfl(myidx, f, 32);
            const float* rp = emb + (size_t)idx * EMB_DIM + lane * 8;
            v4f x0 = *(const v4f*)(rp);        // global_load_b128
            v4f x1 = *(const v4f*)(rp + 4);    // global_load_b128
            a0 += x0;
            a1 += x1;
        }
        float* s = &sh_acc[rl * ACC_STRIDE + lane * 8];
        *(v4f*)(s)     = a0;                   // ds_store_b128
        *(v4f*)(s + 4) = a1;
    }
    __syncthreads();

    const int l15 = lane & 15;
    const int kh  = lane >> 4;   // K-half select per ISA 16x4 A-fragment layout

    // ---------------- Phase 2: layer 1  [32,256] x [256,32]  (f32 WMMA) ----------------
    // Waves 0..3 cover the 2x2 grid of 16x16 output tiles; 64 K-steps of
    // V_WMMA_F32_16X16X4_F32 keep full f32 precision (matches reference dtype).
    if (wave < 4) {
        const int mtile = wave & 1;
        const int ntile = wave >> 1;
        const int arow  = mtile * 16 + l15;    // A-matrix row (batch row in tile)
        const int bcol  = ntile * 16 + l15;    // B-matrix col (output neuron)
        v8f c = {};
        #pragma unroll 8
        for (int k0 = 0; k0 < EMB_DIM; k0 += 4) {
            const int k = k0 + 2 * kh;
            v2f a, bf;
            a.x  = sh_acc[arow * ACC_STRIDE + k];      // padded stride -> 16 distinct banks
            a.y  = sh_acc[arow * ACC_STRIDE + k + 1];
            bf.x = w1[bcol * EMB_DIM + k];             // 32 KB weight, L2-resident
            bf.y = w1[bcol * EMB_DIM + k + 1];
            c = __builtin_amdgcn_wmma_f32_16x16x4_f32(
                    false, a, false, bf, (short)0, c, false, false);
        }
        const float bias = b1[bcol];
        #pragma unroll
        for (int r = 0; r < 8; ++r) {
            // C/D layout: VGPR r, lanes 0-15 -> M=r ; lanes 16-31 -> M=8+r ; N = l15
            const int orow = mtile * 16 + kh * 8 + r;
            float v = c[r] + bias;
            v = v > 0.f ? v : 0.f;                     // ReLU
            sh_h1[orow * H_STRIDE + bcol] = v;
        }
    }
    __syncthreads();

    // ---------------- Phase 3: layer 2  [32,32] x [32,32]  (f32 WMMA) ----------------
    if (wave < 4) {
        const int mtile = wave & 1;
        const int ntile = wave >> 1;
        const int arow  = mtile * 16 + l15;
        const int bcol  = ntile * 16 + l15;
        v8f c = {};
        #pragma unroll
        for (int k0 = 0; k0 < 32; k0 += 4) {
            const int k = k0 + 2 * kh;
            v2f a, bf;
            a.x  = sh_h1[arow * H_STRIDE + k];
            a.y  = sh_h1[arow * H_STRIDE + k + 1];
            bf.x = w2[bcol * 32 + k];
            bf.y = w2[bcol * 32 + k + 1];
            c = __builtin_amdgcn_wmma_f32_16x16x4_f32(
                    false, a, false, bf, (short)0, c, false, false);
        }
        const float bias = b2[bcol];
        #pragma unroll
        for (int r = 0; r < 8; ++r) {
            const int orow = mtile * 16 + kh * 8 + r;
            float v = c[r] + bias;
            v = v > 0.f ? v : 0.f;
            sh_h2[orow * H_STRIDE + bcol] = v;
        }
    }
    __syncthreads();

    // ---------------- Phase 4: layer 3 + tanh (one lane per batch row) ----------------
    if (tid < ROWS) {
        float sum = 0.f;
        #pragma unroll
        for (int j = 0; j < 32; ++j)
            sum += sh_h2[tid * H_STRIDE + j] * w3[j];  // w3 uniform -> s_load, cached
        out[row0 + tid] = tanhf(sum + b3[0]);
    }
}

extern "C" void kernel_launch(void* const* d_in, const int* in_sizes, int n_in,
                              void* d_out, int out_size, void* d_ws, size_t ws_size,
                              hipStream_t stream) {
    const int*   indices = (const int*)  d_in[0];
    const float* emb     = (const float*)d_in[1];
    const float* w1      = (const float*)d_in[2];
    const float* b1      = (const float*)d_in[3];
    const float* w2      = (const float*)d_in[4];
    const float* b2      = (const float*)d_in[5];
    const float* w3      = (const float*)d_in[6];
    const float* b3      = (const float*)d_in[7];
    float*       out     = (float*)d_out;

    const int batch  = in_sizes[0] / N_FEAT;   // indices is [N_FEAT, BATCH]
    const int blocks = batch / ROWS;           // BATCH=16384 -> 512 blocks

    nnue_fused_kernel<<<blocks, 256, 0, stream>>>(
        indices, emb, w1, b1, w2, b2, w3, b3, out, batch);
}